// ConvM_Layer_41377714929840
// MI455X (gfx1250) — compile-verified
//
#include <hip/hip_runtime.h>

typedef float v2f __attribute__((ext_vector_type(2)));
typedef float v8f __attribute__((ext_vector_type(8)));

#define TDIM 4
#define WQ   75
#define WSUP 25
#define CDIM 640
#define HW   100
#define WAY  5
#define SHOT 5
#define NRT  7        // row tiles of 16 covering hw=100 (padded to 112)
#define KS   160      // k-steps of 4 over CDIM=640
#define JT   40       // 16-wide column tiles over CDIM

// ---------------------------------------------------------------------------
// Kernel 1: support per-channel means over shot*hw = 500 samples
// ---------------------------------------------------------------------------
__global__ void smean_kernel(const float* __restrict__ supp,
                             float* __restrict__ smean) {
    int tw = blockIdx.x;                 // 0..19
    int t = tw / WAY, w = tw % WAY;
    for (int c = threadIdx.x; c < CDIM; c += blockDim.x) {
        float s = 0.f;
        for (int sh = 0; sh < SHOT; ++sh) {
            const float* p = supp + (((size_t)(t * WSUP + w * SHOT + sh) * CDIM + c) * HW);
            for (int n = 0; n < HW; n += 4) {
                float4 v = *(const float4*)(p + n);
                s += v.x + v.y + v.z + v.w;
            }
        }
        smean[(size_t)tw * CDIM + c] = s * (1.0f / (SHOT * HW));
    }
}

// ---------------------------------------------------------------------------
// Kernel 2: query per-channel means over hw = 100
// ---------------------------------------------------------------------------
__global__ void qmean_kernel(const float* __restrict__ q,
                             float* __restrict__ qmean) {
    int tq = blockIdx.x;                 // 0..299
    for (int c = threadIdx.x; c < CDIM; c += blockDim.x) {
        const float* p = q + ((size_t)tq * CDIM + c) * HW;
        float s = 0.f;
        for (int n = 0; n < HW; n += 4) {
            float4 v = *(const float4*)(p + n);
            s += v.x + v.y + v.z + v.w;
        }
        qmean[(size_t)tq * CDIM + c] = s * (1.0f / HW);
    }
}

// ---------------------------------------------------------------------------
// Kernel 3: cov build via V_WMMA_F32_16X16X4_F32, one wave per 16x16 tile
// (20*40*40 waves). Output written directly in the *B-operand swizzled*
// layout consumed by the score kernel:
//   covsw[ ((tw*JT + j)*KS + ks)*64 + laneC*2 + vg ]
//     = Cov[k = ks*4 + (laneC>=16 ? 2:0) + vg , d = j*16 + laneC%16] / 99
// ---------------------------------------------------------------------------
__global__ __launch_bounds__(256)
void cov_kernel(const float* __restrict__ supp,
                const float* __restrict__ smean,
                float* __restrict__ covsw) {
    int wave = (blockIdx.x * blockDim.x + threadIdx.x) >> 5;  // 0..31999
    int lane = threadIdx.x & 31;
    int tw  = wave / (JT * JT);
    int rem = wave % (JT * JT);
    int it  = rem / JT;                 // c row-tile (K at consumption)
    int jt  = rem % JT;                 // d col-tile
    int t = tw / WAY, w = tw % WAY;

    int half = lane >> 4;               // 0: K={0,1}, 1: K={2,3}
    int l16  = lane & 15;
    int crow = it * 16 + l16;           // A M-index
    int dcol = jt * 16 + l16;           // B N-index
    float ma = smean[(size_t)tw * CDIM + crow];
    float mb = smean[(size_t)tw * CDIM + dcol];

    const float* baseA = supp + ((size_t)(t * WSUP + w * SHOT) * CDIM + crow) * HW;
    const float* baseB = supp + ((size_t)(t * WSUP + w * SHOT) * CDIM + dcol) * HW;

    v8f acc = {};
    for (int s = 0; s < SHOT; ++s) {
        const float* pA = baseA + (size_t)s * CDIM * HW + half * 2;
        const float* pB = baseB + (size_t)s * CDIM * HW + half * 2;
#pragma unroll 5
        for (int nn = 0; nn < HW; nn += 4) {
            v2f a, b;
            a.x = pA[nn]     - ma;
            a.y = pA[nn + 1] - ma;
            b.x = pB[nn]     - mb;
            b.y = pB[nn + 1] - mb;
            acc = __builtin_amdgcn_wmma_f32_16x16x4_f32(
                false, a, false, b, (short)0, acc, false, false);
        }
    }
    const float scale = 1.0f / (HW - 1);
    // D layout: VGPR v, lanes 0-15 -> M=v; lanes 16-31 -> M=v+8; N = lane%16
#pragma unroll
    for (int v = 0; v < 8; ++v) {
        int c_local = v + 8 * half;                   // K index within tile
        int ks      = it * 4 + (c_local >> 2);
        int kk      = c_local & 3;
        int laneC   = (kk >> 1) * 16 + l16;           // consumer lane
        int vg      = kk & 1;                         // consumer vgpr
        covsw[(((size_t)tw * JT + jt) * KS + ks) * 64 + laneC * 2 + vg] = acc[v] * scale;
    }
}

// ---------------------------------------------------------------------------
// Kernel 4 (fast path only): centered query panels in *A-operand swizzled*
// layout:  qsw[ ((tq*NRT + rt)*KS + ks)*64 + lane*2 + vg ]
//            = Qbar[n = rt*16 + lane%16 , k = ks*4 + (lane>=16?2:0) + vg]
// Pad rows (n >= 100) are written as zeros.
// ---------------------------------------------------------------------------
__global__ void qswizzle_kernel(const float* __restrict__ q,
                                const float* __restrict__ qmean,
                                float* __restrict__ qsw) {
    int blk = blockIdx.x;               // tq*NRT + rt
    int tq = blk / NRT, rt = blk % NRT;
    const float* qb = q + (size_t)tq * CDIM * HW;   // layout [c][n]
    const float* qm = qmean + (size_t)tq * CDIM;
    float* outb = qsw + (size_t)blk * KS * 64;
    for (int p = threadIdx.x; p < KS * 32; p += blockDim.x) {
        int ks = p >> 5, lane = p & 31;
        int half = lane >> 4, l16 = lane & 15;
        int n  = rt * 16 + l16;
        int k0 = ks * 4 + half * 2;
        bool val = n < HW;
        int nc = val ? n : 0;
        float a0 = qb[(size_t)k0 * HW + nc]       - qm[k0];
        float a1 = qb[(size_t)(k0 + 1) * HW + nc] - qm[k0 + 1];
        v2f a;
        a.x = val ? a0 : 0.f;
        a.y = val ? a1 : 0.f;
        *(v2f*)(outb + (size_t)p * 2) = a;
    }
}

// ---------------------------------------------------------------------------
// Kernel 5: score[t,q,w] = bias + sum_n convw[n]*leaky( q_n^T Cov q_n )
// One 224-thread block (7 waves) per (t,q,w); wave r owns rows 16r..16r+15.
// FAST: hot loop = 1 b64 (A swizzled) + 1 b64 (B swizzled) + 1 wmma.
// MID : A loaded directly from query (scattered), B swizzled.
// ---------------------------------------------------------------------------
template <bool FAST>
__global__ __launch_bounds__(224)
void score_kernel(const float* __restrict__ q,
                  const float* __restrict__ qmean,
                  const float* __restrict__ covsw,
                  const float* __restrict__ qsw,
                  const float* __restrict__ convw,
                  const float* __restrict__ convb,
                  float* __restrict__ out) {
    int blk = blockIdx.x;               // (t*WQ + q)*WAY + w
    int w   = blk % WAY;
    int tq  = blk / WAY;
    int t   = tq / WQ;
    int wv   = threadIdx.x >> 5;        // 0..6 row tile
    int lane = threadIdx.x & 31;
    int half = lane >> 4;
    int l16  = lane & 15;

    int tw = t * WAY + w;
    const float* qb = q + (size_t)tq * CDIM * HW;   // layout [c][n]
    const float* qm = qmean + (size_t)tq * CDIM;
    const float* qswp  = qsw + (size_t)(tq * NRT + wv) * KS * 64;
    const float* paF   = qswp + lane * 2;

    // MID-path A addressing
    int arow   = wv * 16 + l16;
    bool aval  = arow < HW;
    int arow_c = aval ? arow : 0;
    const float* paM = qb + arow_c;

    int d = 0;  // per-lane B column, set per j
    float partial[8];
#pragma unroll
    for (int v = 0; v < 8; ++v) partial[v] = 0.f;

    for (int j = 0; j < JT; ++j) {
        d = j * 16 + l16;
        const float* pb = covsw + (((size_t)tw * JT + j) * KS) * 64 + lane * 2;
        v8f acc = {};
        if (FAST) {
#pragma unroll 8
            for (int ks = 0; ks < KS; ++ks) {
                v2f a = *(const v2f*)(paF + (size_t)ks * 64);
                v2f b = *(const v2f*)(pb  + (size_t)ks * 64);
                acc = __builtin_amdgcn_wmma_f32_16x16x4_f32(
                    false, a, false, b, (short)0, acc, false, false);
            }
        } else {
#pragma unroll 4
            for (int ks = 0; ks < KS; ++ks) {
                int k = ks * 4 + half * 2;
                float a0 = paM[(size_t)k * HW]      - qm[k];
                float a1 = paM[(size_t)k * HW + HW] - qm[k + 1];
                a0 = aval ? a0 : 0.f;
                a1 = aval ? a1 : 0.f;
                v2f a; a.x = a0; a.y = a1;
                v2f b = *(const v2f*)(pb + (size_t)ks * 64);
                acc = __builtin_amdgcn_wmma_f32_16x16x4_f32(
                    false, a, false, b, (short)0, acc, false, false);
            }
        }
        // multiply QC tile elementwise by Qbar tile, fold into row partials
        if (FAST) {
            int dks = d >> 2;
            int dh  = (d & 3) >> 1;
            int vg  = d & 1;
            const float* qt = qswp + (size_t)dks * 64 + dh * 32 + vg;
#pragma unroll
            for (int v = 0; v < 8; ++v) {
                int rl = v + 8 * half;                 // row % 16
                float qv = qt[rl * 2];                 // already centered / zero-padded
                partial[v] += acc[v] * qv;
            }
        } else {
#pragma unroll
            for (int v = 0; v < 8; ++v) {
                int row = wv * 16 + v + 8 * half;
                int rc  = row < HW ? row : 0;
                float qv = qb[(size_t)d * HW + rc] - qm[d];
                qv = row < HW ? qv : 0.f;
                partial[v] += acc[v] * qv;
            }
        }
    }

    // Reduce each row over its 16 N-lanes, scale, LeakyReLU(0.2), conv weight.
    const float scale = 1.0f / (HW - 1);
    float total = 0.f;
#pragma unroll
    for (int v = 0; v < 8; ++v) {
        float s = partial[v];
        s += __shfl_xor(s, 1, 32);
        s += __shfl_xor(s, 2, 32);
        s += __shfl_xor(s, 4, 32);
        s += __shfl_xor(s, 8, 32);      // uniform across each 16-lane half
        int row = wv * 16 + v + 8 * half;
        int rc  = row < HW ? row : 0;
        float cs = s * scale;
        float x  = cs >= 0.f ? cs : 0.2f * cs;
        float contrib = x * convw[rc];
        total += (row < HW) ? contrib : 0.f;
    }
    total += __shfl_xor(total, 16, 32); // combine the two halves

    __shared__ float red[NRT];
    if (lane == 0) red[wv] = total;
    __syncthreads();
    if (threadIdx.x == 0) {
        float r = convb[0];
#pragma unroll
        for (int i = 0; i < NRT; ++i) r += red[i];
        out[blk] = r;
    }
}

// ---------------------------------------------------------------------------
extern "C" void kernel_launch(void* const* d_in, const int* in_sizes, int n_in,
                              void* d_out, int out_size, void* d_ws, size_t ws_size,
                              hipStream_t stream) {
    const float* qf = (const float*)d_in[0];   // (4,75,640,10,10)
    const float* sf = (const float*)d_in[1];   // (4,25,640,10,10)
    const float* cw = (const float*)d_in[2];   // (1,1,100)
    const float* cb = (const float*)d_in[3];   // (1,)
    float* out = (float*)d_out;                // (4,75,5)

    const size_t smean_sz = (size_t)TDIM * WAY * CDIM;        //  12,800
    const size_t qmean_sz = (size_t)TDIM * WQ  * CDIM;        // 192,000
    const size_t covsw_sz = (size_t)TDIM * WAY * JT * KS * 64;   // 8,192,000 (32.8 MB)
    const size_t qsw_sz   = (size_t)TDIM * WQ * NRT * KS * 64;   // 21,504,000 (86 MB)

    float* smean = (float*)d_ws;
    float* qmean = smean + smean_sz;
    float* covsw = qmean + qmean_sz;
    float* qsw   = covsw + covsw_sz;

    const size_t need_fast = (smean_sz + qmean_sz + covsw_sz + qsw_sz) * sizeof(float);
    const bool fast = ws_size >= need_fast;

    smean_kernel<<<TDIM * WAY, 256, 0, stream>>>(sf, smean);
    qmean_kernel<<<TDIM * WQ, 256, 0, stream>>>(qf, qmean);

    // 20*40*40 = 32000 waves, 8 waves per block -> 4000 blocks
    cov_kernel<<<4000, 256, 0, stream>>>(sf, smean, covsw);

    if (fast) {
        qswizzle_kernel<<<TDIM * WQ * NRT, 256, 0, stream>>>(qf, qmean, qsw);
        score_kernel<true><<<TDIM * WQ * WAY, 224, 0, stream>>>(
            qf, qmean, covsw, qsw, cw, cb, out);
    } else {
        score_kernel<false><<<TDIM * WQ * WAY, 224, 0, stream>>>(
            qf, qmean, covsw, qsw, cw, cb, out);
    }
}